// H2Conv_87205015978220
// MI455X (gfx1250) — compile-verified
//
#include <hip/hip_runtime.h>

typedef __attribute__((ext_vector_type(2))) float v2f;
typedef __attribute__((ext_vector_type(4))) float v4f;
typedef __attribute__((ext_vector_type(8))) float v8f;

constexpr int N_NODES = 100000;
constexpr int E_EDGES = 20000;
constexpr long NNZ    = 1000000;
constexpr int IN_CH   = 128;
constexpr int D       = 128;          // HEADS * OUT_CH
constexpr int LDSTRIDE = 132;         // 132 % 64 banks == 4 -> conflict-free b64 reads

// ---------------------------------------------------------------------------
// Kernel 1: Xh = lorentz_linear(X @ W^T + b)   fused GEMM + rowwise epilogue.
// One wave32 computes a 16x128 output tile using v_wmma_f32_16x16x4_f32.
// C layout (32-bit 16x16): lanes 0-15 -> rows j (VGPR j), N=lane;
//                          lanes 16-31 -> rows j+8, N=lane-16.
// A layout (32-bit 16x4):  lanes 0-15 hold K=0,1; lanes 16-31 hold K=2,3.
// ---------------------------------------------------------------------------
__global__ __launch_bounds__(128) void gemm_lorentz_kernel(
    const float* __restrict__ X, const float* __restrict__ W,
    const float* __restrict__ bvec, const float* __restrict__ scale_log,
    float* __restrict__ Xh, float* __restrict__ out)
{
    __shared__ float Wlds[D * LDSTRIDE];

    const int tid = threadIdx.x;

    // Stage all of W (128x128 fp32 = 64 KB) into LDS with padded stride.
    for (int it = tid; it < D * (IN_CH / 4); it += 128) {
        int r  = it >> 5;            // row of W (D dimension)
        int k4 = (it & 31) * 4;      // K offset, multiple of 4
        v4f w = *reinterpret_cast<const v4f*>(W + (long)r * IN_CH + k4);
        *reinterpret_cast<v4f*>(&Wlds[r * LDSTRIDE + k4]) = w;
    }
    __syncthreads();

    const int lane = tid & 31;
    const int wave = tid >> 5;
    const int tile = blockIdx.x * 4 + wave;   // 16-row output tile index
    if (tile * 16 >= N_NODES) return;         // whole wave exits together

    const int halfrow = lane & 15;            // M (for A) / N (for B,C) in-tile
    const int khalf   = (lane >> 4) * 2;      // K sub-offset: 0 or 2

    const float* Arow = X + (long)(tile * 16 + halfrow) * IN_CH + khalf;

    v8f acc[8];
    #pragma unroll
    for (int nt = 0; nt < 8; ++nt)
        #pragma unroll
        for (int j = 0; j < 8; ++j) acc[nt][j] = 0.0f;

    // K loop: 32 steps of K=4, 8 N-tiles -> 256 WMMAs per wave.
    for (int kk = 0; kk < IN_CH; kk += 4) {
        v2f a = *reinterpret_cast<const v2f*>(Arow + kk);
        #pragma unroll
        for (int nt = 0; nt < 8; ++nt) {
            v2f bb = *reinterpret_cast<const v2f*>(
                &Wlds[(nt * 16 + halfrow) * LDSTRIDE + kk + khalf]);
            acc[nt] = __builtin_amdgcn_wmma_f32_16x16x4_f32(
                /*neg_a=*/false, a, /*neg_b=*/false, bb,
                /*c_mod=*/(short)0, acc[nt],
                /*reuse_a=*/false, /*reuse_b=*/false);
        }
    }

    // Bias add (same bias across the 8 rows each lane holds per tile).
    #pragma unroll
    for (int nt = 0; nt < 8; ++nt) {
        float bval = bvec[nt * 16 + halfrow];
        #pragma unroll
        for (int j = 0; j < 8; ++j) acc[nt][j] += bval;
    }

    const float es      = __expf(scale_log[0]);
    const int   srcLane = lane & 16;              // broadcast source per half
    const int   rowbase = tile * 16 + (lane >> 4) * 8;

    #pragma unroll
    for (int j = 0; j < 8; ++j) {
        // sum of squares over columns 1..127 for this row
        float s = 0.0f;
        #pragma unroll
        for (int nt = 0; nt < 8; ++nt) { float v = acc[nt][j]; s += v * v; }
        float col0 = acc[0][j];                   // column 0 lives at halfrow==0
        if (halfrow == 0) s -= col0 * col0;
        #pragma unroll
        for (int m = 1; m < 16; m <<= 1) s += __shfl_xor(s, m, 32);

        float y0    = __shfl(col0, srcLane, 32);  // broadcast y[row,0]
        float timev = es / (1.0f + __expf(-y0)) + 1.1f;
        float rs    = sqrtf((timev * timev - 1.0f) / (s + 1e-8f));

        const int row = rowbase + j;
        #pragma unroll
        for (int nt = 0; nt < 8; ++nt) {
            float v = acc[nt][j] * rs;
            if (nt == 0 && halfrow == 0) v = timev;   // time coordinate
            long idx = (long)row * D + nt * 16 + halfrow;
            Xh[idx]  = v;   // workspace copy for the gathers
            out[idx] = v;   // out starts at Xh; vertex scatter adds eps*Xe on top
        }
    }
}

// ---------------------------------------------------------------------------
// Kernel 0: zero the Xe accumulator (E x 128).
// ---------------------------------------------------------------------------
__global__ void zero_kernel(float* __restrict__ p, long n)
{
    long i = (long)blockIdx.x * blockDim.x + threadIdx.x;
    if (i < n) p[i] = 0.0f;
}

// ---------------------------------------------------------------------------
// Kernel 2: Xe[e] += Xh[v] - emb_ty[t]  (one thread per incidence-channel).
// ---------------------------------------------------------------------------
__global__ void scatter_edges_kernel(
    const float* __restrict__ Xh, const float* __restrict__ emb,
    const int* __restrict__ vertex, const int* __restrict__ edges,
    const int* __restrict__ types, float* __restrict__ Xe)
{
    long idx = (long)blockIdx.x * blockDim.x + threadIdx.x;
    long i = idx >> 7;
    int  c = (int)(idx & 127);
    if (i >= NNZ) return;
    int v = vertex[i], e = edges[i], t = types[i];
    float val = Xh[(long)v * D + c] - emb[(long)t * D + c];
    atomicAdd(&Xe[(long)e * D + c], val);
}

// ---------------------------------------------------------------------------
// Kernel 3: out[v] += eps * Xe[e]  (out pre-initialized to Xh by kernel 1).
// ---------------------------------------------------------------------------
__global__ void scatter_verts_kernel(
    const float* __restrict__ Xe,
    const int* __restrict__ vertex, const int* __restrict__ edges,
    const float* __restrict__ eps, float* __restrict__ out)
{
    long idx = (long)blockIdx.x * blockDim.x + threadIdx.x;
    long i = idx >> 7;
    int  c = (int)(idx & 127);
    if (i >= NNZ) return;
    float epsv = eps[0];
    int v = vertex[i], e = edges[i];
    atomicAdd(&out[(long)v * D + c], epsv * Xe[(long)e * D + c]);
}

extern "C" void kernel_launch(void* const* d_in, const int* in_sizes, int n_in,
                              void* d_out, int out_size, void* d_ws, size_t ws_size,
                              hipStream_t stream)
{
    const float* X         = (const float*)d_in[0];
    const float* emb       = (const float*)d_in[1];
    const float* W         = (const float*)d_in[2];
    const float* b         = (const float*)d_in[3];
    const float* scale_log = (const float*)d_in[4];
    const float* eps       = (const float*)d_in[5];
    const int*   vertex    = (const int*)d_in[6];
    const int*   edges     = (const int*)d_in[7];
    const int*   types     = (const int*)d_in[8];

    float* Xh  = (float*)d_ws;                       // N x 128  (51.2 MB)
    float* Xe  = Xh + (size_t)N_NODES * D;           // E x 128  (10.2 MB)
    float* out = (float*)d_out;

    long ne = (long)E_EDGES * D;
    zero_kernel<<<(int)((ne + 255) / 256), 256, 0, stream>>>(Xe, ne);

    int ntiles  = N_NODES / 16;                      // 6250
    int nblocks = (ntiles + 3) / 4;                  // 4 waves (tiles) per block
    gemm_lorentz_kernel<<<nblocks, 128, 0, stream>>>(X, W, b, scale_log, Xh, out);

    long total = NNZ * (long)D;
    int  sblocks = (int)((total + 255) / 256);
    scatter_edges_kernel<<<sblocks, 256, 0, stream>>>(Xh, emb, vertex, edges, types, Xe);
    scatter_verts_kernel<<<sblocks, 256, 0, stream>>>(Xe, vertex, edges, eps, out);
}